// N3Block_15874199126625
// MI455X (gfx1250) — compile-verified
//
#include <hip/hip_runtime.h>

// ---------------------------------------------------------------------------
// Problem constants (from the reference)
// ---------------------------------------------------------------------------
constexpr int B_ = 4, C_ = 8, H_ = 240, W_ = 240, F_ = 64;
constexpr int P_ = 10, S_ = 5, O_ = 15, K_ = 7;
constexpr int G_ = (H_ - P_) / S_ + 1;          // 47
constexpr int M_ = G_ * G_;                     // 2209 patches
constexpr int O2_ = O_ * O_;                    // 225 neighbors
constexpr int HW_ = H_ * W_;
constexpr int DG_ = C_ * P_ * P_;               // 800 patch dim
constexpr float EPS_ = 1e-5f;
constexpr float NEG_ = -1e9f;

typedef __attribute__((ext_vector_type(2))) float v2f;
typedef __attribute__((ext_vector_type(8))) float v8f;

__device__ __forceinline__ int clipw(int v) {   // neighbor-window start clip
  return min(max(v - O_ / 2, 0), G_ - O_);
}

// ---------------------------------------------------------------------------
// conv1: 8 -> 64, 3x3 SAME, fused BN + ReLU (direct; K=72 too small for MMA)
// ---------------------------------------------------------------------------
__global__ __launch_bounds__(256)
void k_conv1(const float* __restrict__ x, const float* __restrict__ w,
             const float* __restrict__ gg, const float* __restrict__ bb,
             const float* __restrict__ mmu, const float* __restrict__ vvr,
             float* __restrict__ out)
{
  int idx = blockIdx.x * 256 + threadIdx.x;
  if (idx >= B_ * F_ * HW_) return;
  int px = idx % W_, py = (idx / W_) % H_;
  int f = (idx / HW_) % F_, b = idx / (F_ * HW_);
  float acc = 0.f;
  #pragma unroll
  for (int ky = 0; ky < 3; ++ky) {
    int yy = py + ky - 1; if (yy < 0 || yy >= H_) continue;
    #pragma unroll
    for (int kx = 0; kx < 3; ++kx) {
      int xx = px + kx - 1; if (xx < 0 || xx >= W_) continue;
      const float* xp = x + (size_t)(b * C_) * HW_ + (size_t)yy * W_ + xx;
      const float* wp = w + (size_t)(f * C_) * 9 + ky * 3 + kx;
      #pragma unroll
      for (int c = 0; c < C_; ++c) acc += xp[(size_t)c * HW_] * wp[c * 9];
    }
  }
  float s = gg[f] * rsqrtf(vvr[f] + EPS_);
  float o = acc * s + (bb[f] - mmu[f] * s);
  out[idx] = fmaxf(o, 0.f);
}

// ---------------------------------------------------------------------------
// Pack conv2 weights (OIHW) into WMMA f32 16x16x4 B-fragment order so each
// lane's (b0,b1) pair is one coalesced 8-byte load.
// ---------------------------------------------------------------------------
__global__ __launch_bounds__(256)
void k_packw(const float* __restrict__ w, float* __restrict__ wp)
{
  int idx = blockIdx.x * 256 + threadIdx.x;
  if (idx >= 9 * 16 * 4 * 64) return;          // 36864
  int t    = idx & 1;
  int lane = (idx >> 1) & 31;
  int ng   = (idx >> 6) & 3;
  int cq   = (idx >> 8) & 15;
  int kk   = idx >> 12;
  int half = lane >> 4, n = lane & 15;
  int o  = ng * 16 + n;
  int ci = cq * 4 + 2 * half + t;
  wp[idx] = w[(size_t)(o * F_ + ci) * 9 + kk];
}

// ---------------------------------------------------------------------------
// conv2: 64 -> 64, 3x3 SAME, fused BN+ReLU as implicit GEMM on
// V_WMMA_F32_16X16X4_F32 (validated in round 1: clean wmma + s_clause'd loads)
// ---------------------------------------------------------------------------
__global__ __launch_bounds__(128)
void k_conv2(const float* __restrict__ in, const float* __restrict__ w2,
             const float* __restrict__ wpack,
             const float* __restrict__ gg, const float* __restrict__ bb,
             const float* __restrict__ mmu, const float* __restrict__ vvr,
             float* __restrict__ out)
{
#if defined(__HIP_DEVICE_COMPILE__) && __has_builtin(__builtin_amdgcn_wmma_f32_16x16x4_f32)
  __shared__ float smem[4][16][17];            // [wave][out-ch][pixel], padded
  const int tid = threadIdx.x;
  const int wv = tid >> 5, lane = tid & 31;
  const int half = lane >> 4, mr = lane & 15;
  const int xt = blockIdx.x % (W_ / 16);
  const int y  = (blockIdx.x / (W_ / 16)) % H_;
  const int b  = blockIdx.x / ((W_ / 16) * H_);
  const int x0 = xt * 16;
  const float* inb = in + (size_t)b * F_ * HW_;
  v8f acc = {};
  for (int ky = 0; ky < 3; ++ky) {
    int yy = y + ky - 1;
    bool yok = ((unsigned)yy < (unsigned)H_);
    int yc = yok ? yy : 0;
    for (int kx = 0; kx < 3; ++kx) {
      int xa = x0 + mr + kx - 1;
      bool ok = yok && ((unsigned)xa < (unsigned)W_);
      int xc = ok ? xa : 0;
      const float* ab = inb + (size_t)yc * W_ + xc;
      const float msk = ok ? 1.f : 0.f;
      const float* bb0 = wpack + (size_t)(((ky * 3 + kx) * 16) * 4 + wv) * 64 + 2 * lane;
      #pragma unroll
      for (int cq = 0; cq < 16; ++cq) {
        const int c0 = cq * 4 + 2 * half;
        v2f a;
        a.x = ab[(size_t)c0 * HW_] * msk;
        a.y = ab[(size_t)(c0 + 1) * HW_] * msk;
        const float* bp = bb0 + (size_t)cq * 256;
        v2f bf; bf.x = bp[0]; bf.y = bp[1];
        acc = __builtin_amdgcn_wmma_f32_16x16x4_f32(
            false, a, false, bf, (short)0, acc, false, false);
      }
    }
  }
  #pragma unroll
  for (int r = 0; r < 8; ++r) smem[wv][mr][r + 8 * half] = acc[r];
  __syncthreads();
  #pragma unroll
  for (int it = 0; it < 8; ++it) {
    int ch = it * 2 + half;
    float v = smem[wv][ch][mr];
    int cg = wv * 16 + ch;
    float s = gg[cg] * rsqrtf(vvr[cg] + EPS_);
    float o = v * s + (bb[cg] - mmu[cg] * s);
    out[((size_t)(b * F_ + cg) * H_ + y) * W_ + x0 + mr] = fmaxf(o, 0.f);
  }
  (void)w2;
#else
  const int tid = threadIdx.x;
  const int xt = blockIdx.x % (W_ / 16);
  const int y  = (blockIdx.x / (W_ / 16)) % H_;
  const int b  = blockIdx.x / ((W_ / 16) * H_);
  const int x0 = xt * 16;
  const float* inb = in + (size_t)b * F_ * HW_;
  for (int q = 0; q < 8; ++q) {
    int o = tid + q * 128;
    int ch = o >> 4, pxi = o & 15;
    int xx0 = x0 + pxi;
    float acc = 0.f;
    for (int ky = 0; ky < 3; ++ky) {
      int yy = y + ky - 1; if (yy < 0 || yy >= H_) continue;
      for (int kx = 0; kx < 3; ++kx) {
        int xx = xx0 + kx - 1; if (xx < 0 || xx >= W_) continue;
        const float* ip = inb + (size_t)yy * W_ + xx;
        const float* wp = w2 + (size_t)(ch * F_) * 9 + ky * 3 + kx;
        for (int c = 0; c < F_; ++c) acc += ip[(size_t)c * HW_] * wp[c * 9];
      }
    }
    float s = gg[ch] * rsqrtf(vvr[ch] + EPS_);
    float ov = acc * s + (bb[ch] - mmu[ch] * s);
    out[((size_t)(b * F_ + ch) * H_ + y) * W_ + xx0] = fmaxf(ov, 0.f);
  }
  (void)wpack;
#endif
}

// ---------------------------------------------------------------------------
// conv3: 64 -> Cout, 3x3 SAME, + bias (Cout = 8 or 1)
// ---------------------------------------------------------------------------
__global__ __launch_bounds__(256)
void k_conv3(const float* __restrict__ in, const float* __restrict__ w,
             const float* __restrict__ bias, float* __restrict__ out, int Cout)
{
  int idx = blockIdx.x * 256 + threadIdx.x;
  if (idx >= B_ * Cout * HW_) return;
  int px = idx % W_, py = (idx / W_) % H_;
  int co = (idx / HW_) % Cout, b = idx / (Cout * HW_);
  float acc = bias[co];
  const float* inb = in + (size_t)b * F_ * HW_;
  for (int ky = 0; ky < 3; ++ky) {
    int yy = py + ky - 1; if (yy < 0 || yy >= H_) continue;
    for (int kx = 0; kx < 3; ++kx) {
      int xx = px + kx - 1; if (xx < 0 || xx >= W_) continue;
      const float* ip = inb + (size_t)yy * W_ + xx;
      const float* wp = w + (size_t)(co * F_) * 9 + ky * 3 + kx;
      for (int c = 0; c < F_; ++c) acc += ip[(size_t)c * HW_] * wp[c * 9];
    }
  }
  out[idx] = acc;
}

// ---------------------------------------------------------------------------
// Per-patch stats: ynorm[b,m] = sum(xe_patch^2); ltm[b,m] = mean(log_t patch).
// One wave32 per (b,m).
// ---------------------------------------------------------------------------
__global__ __launch_bounds__(256)
void k_stats(const float* __restrict__ xe, const float* __restrict__ ltimg,
             float* __restrict__ ynorm, float* __restrict__ ltm)
{
  int gtid = blockIdx.x * 256 + threadIdx.x;
  int wid = gtid >> 5;
  int lane = threadIdx.x & 31;
  if (wid >= B_ * M_) return;
  int m = wid % M_, b = wid / M_;
  int gy = m / G_, gx = m % G_;
  int y0 = gy * S_, x0 = gx * S_;
  float s = 0.f;
  for (int d = lane; d < DG_; d += 32) {
    int ch = d / 100, r = d % 100, pi = r / 10, pj = r % 10;
    float v = xe[((size_t)(b * C_ + ch) * H_ + y0 + pi) * W_ + x0 + pj];
    s += v * v;
  }
  #pragma unroll
  for (int off = 16; off > 0; off >>= 1) s += __shfl_xor(s, off, 32);
  float t = 0.f;
  for (int d = lane; d < 100; d += 32) {
    int pi = d / 10, pj = d % 10;
    t += ltimg[((size_t)b * H_ + y0 + pi) * W_ + x0 + pj];
  }
  #pragma unroll
  for (int off = 16; off > 0; off >>= 1) t += __shfl_xor(t, off, 32);
  if (lane == 0) { ynorm[wid] = s; ltm[wid] = t * 0.01f; }
}

// ---------------------------------------------------------------------------
// Distances -> logits as a banded correlation GEMM on WMMA f32 16x16x4.
// Block = (b, gy, gx-tile of 16, cy); 64 threads = 2 waves, wave = nt (which
// 16-wide slab of the 32 neighbor columns wx in [wx_lo, wx_lo+31]).
// GEMM: rows = 16 patches of the strip, cols = neighbor patches at (wy, wx),
// contraction over the 800-dim embedded patch. Scatter applies the per-row
// mapping cx = wx - wxs(m) and the logits formula; ragged/out-of-band entries
// are rejected by the bounds checks (their loads are address-clamped only).
// ---------------------------------------------------------------------------
__global__ __launch_bounds__(64)
void k_distw(const float* __restrict__ xe, const float* __restrict__ ynorm,
             const float* __restrict__ ltm, float* __restrict__ logits)
{
  int blk = blockIdx.x;
  int cy  = blk % O_;
  int gxt = (blk / O_) % 3;
  int gy  = (blk / (3 * O_)) % G_;
  int b   = blk / (3 * O_ * G_);
  int gx0 = gxt * 16;
  int width = min(16, G_ - gx0);
  int wys = clipw(gy);
  int wx_lo = clipw(gx0);
  int wy = wys + cy;
  const float* xeb = xe + (size_t)(b * C_) * HW_;
#if defined(__HIP_DEVICE_COMPILE__) && __has_builtin(__builtin_amdgcn_wmma_f32_16x16x4_f32)
  const int nt = threadIdx.x >> 5;             // wave id = column slab
  const int lane = threadIdx.x & 31;
  const int half = lane >> 4, ln = lane & 15;
  const int gxa = min(gx0 + ln, G_ - 1);       // A row source (clamped)
  const int wx  = wx_lo + nt * 16 + ln;        // B column (logical)
  const int wxc = min(wx, G_ - 1);             // B column load address (clamped)
  v8f acc = {};
  for (int kc = 0; kc < DG_ / 4; ++kc) {       // 200 k-steps of 4
    int k0 = kc * 4 + 2 * half;                // k0 even -> (k0,k0+1) same row
    int ch = k0 / 100; int rem = k0 - ch * 100;
    int pi = rem / 10;  int pj = rem - pi * 10;
    const float* pl = xeb + (size_t)ch * HW_ + (size_t)pi * W_ + pj;
    const float* ar = pl + (size_t)(gy * S_) * W_ + gxa * S_;
    const float* br = pl + (size_t)(wy * S_) * W_ + wxc * S_;
    v2f a;  a.x  = ar[0]; a.y  = ar[1];
    v2f bf; bf.x = br[0]; bf.y = br[1];
    acc = __builtin_amdgcn_wmma_f32_16x16x4_f32(
        false, a, false, bf, (short)0, acc, false, false);
  }
  float iexp = 0.f;                            // filled lazily per row
  #pragma unroll
  for (int r = 0; r < 8; ++r) {
    int i = r + 8 * half;                      // patch row within strip
    if (i >= width) continue;
    int gx = gx0 + i;
    int cx = wx - clipw(gx);
    if (cx < 0 || cx >= O_) continue;
    int m  = gy * G_ + gx;
    int mp = wy * G_ + wx;
    float D = ynorm[(size_t)b * M_ + m] + ynorm[(size_t)b * M_ + mp] - 2.f * acc[r];
    iexp = expf(-ltm[(size_t)b * M_ + m]);
    logits[((size_t)b * M_ + m) * O2_ + cy * O_ + cx] = (mp == m) ? NEG_ : -D * iexp;
  }
  (void)iexp;
#else
  // host-parse / no-builtin fallback: same coverage, direct dots
  for (int e = threadIdx.x; e < width * O_; e += 64) {
    int i = e / O_, cx = e % O_;
    int gx = gx0 + i;
    int wxs = clipw(gx);
    int wxv = wxs + cx;
    int m = gy * G_ + gx, mp = wy * G_ + wxv;
    float dot = 0.f;
    for (int d = 0; d < DG_; ++d) {
      int ch = d / 100, rem = d % 100, pi = rem / 10, pj = rem % 10;
      const float* pl = xeb + (size_t)ch * HW_ + (size_t)pi * W_ + pj;
      dot += pl[(size_t)(gy * S_) * W_ + gx * S_] * pl[(size_t)(wy * S_) * W_ + wxv * S_];
    }
    float D = ynorm[(size_t)b * M_ + m] + ynorm[(size_t)b * M_ + mp] - 2.f * dot;
    logits[((size_t)b * M_ + m) * O2_ + cy * O_ + cx] =
        (mp == m) ? NEG_ : -D * expf(-ltm[(size_t)b * M_ + m]);
  }
#endif
}

// ---------------------------------------------------------------------------
// K=7 rounds of softmax + log1p(-w) deflation. One wave32 per (b,m).
// ---------------------------------------------------------------------------
__global__ __launch_bounds__(256)
void k_softmax(const float* __restrict__ logits, float* __restrict__ Wk)
{
  int gtid = blockIdx.x * 256 + threadIdx.x;
  int wid = gtid >> 5;
  int lane = threadIdx.x & 31;
  if (wid >= B_ * M_) return;
  float l[8];
  const float* lp = logits + (size_t)wid * O2_;
  #pragma unroll
  for (int j = 0; j < 8; ++j) { int c = lane + 32 * j; l[j] = (c < O2_) ? lp[c] : NEG_; }
  float* wp = Wk + (size_t)wid * K_ * O2_;
  for (int k = 0; k < K_; ++k) {
    float mx = NEG_;
    #pragma unroll
    for (int j = 0; j < 8; ++j) mx = fmaxf(mx, l[j]);
    #pragma unroll
    for (int off = 16; off > 0; off >>= 1) mx = fmaxf(mx, __shfl_xor(mx, off, 32));
    float e[8]; float sum = 0.f;
    #pragma unroll
    for (int j = 0; j < 8; ++j) { e[j] = expf(l[j] - mx); sum += e[j]; }
    #pragma unroll
    for (int off = 16; off > 0; off >>= 1) sum += __shfl_xor(sum, off, 32);
    float inv = 1.f / sum;
    #pragma unroll
    for (int j = 0; j < 8; ++j) {
      int c = lane + 32 * j;
      float w = e[j] * inv;
      if (c < O2_) wp[(size_t)k * O2_ + c] = w;
      l[j] += log1pf(-fminf(w, 1.f - 1e-6f));
    }
  }
}

// ---------------------------------------------------------------------------
// out[:, 0:8] = x  (runs after the memset of d_out)
// ---------------------------------------------------------------------------
__global__ __launch_bounds__(256)
void k_copyx(const float* __restrict__ x, float* __restrict__ out)
{
  int idx = blockIdx.x * 256 + threadIdx.x;
  if (idx >= B_ * C_ * HW_) return;
  int p = idx % HW_; int ch = (idx / HW_) % C_; int b = idx / (C_ * HW_);
  out[(size_t)(b * (C_ + K_ * C_) + ch) * HW_ + p] = x[idx];
}

// ---------------------------------------------------------------------------
// Aggregation on WMMA: per (b,m), Z(7x800) = Wk_row(7x225) * Y(225x800)
// where Y[c][d] is gathered from the raw image (neighbor patch c, element d).
// A (Wk row, 6.3KB) is staged in LDS -> ds-load fragments; K padded 225->228
// with zero A rows; M padded 7->16 (rows 7..15 zero). The 16x16 accumulator
// is scattered with global_atomic_add_f32 into the zeroed output (<=4 adds
// per output element from overlapping patch footprints). Block = 4 waves,
// wave handles d-tiles wv, wv+4, ... (N = 800 = 50 tiles exactly).
// ---------------------------------------------------------------------------
__global__ __launch_bounds__(128)
void k_aggw(const float* __restrict__ x, const float* __restrict__ Wk,
            float* __restrict__ out)
{
  __shared__ float sw[K_ * O2_];               // 1575 floats
  int bm = blockIdx.x;
  int m = bm % M_, b = bm / M_;
  int gy = m / G_, gx = m % G_;
  int wys = clipw(gy), wxs = clipw(gx);
  for (int e = threadIdx.x; e < K_ * O2_; e += 128)
    sw[e] = Wk[(size_t)(b * M_ + m) * K_ * O2_ + e];
  __syncthreads();
  const float* xb = x + (size_t)(b * C_) * HW_;
  float* ob = out + (size_t)(b * (C_ + K_ * C_) + C_) * HW_;
#if defined(__HIP_DEVICE_COMPILE__) && __has_builtin(__builtin_amdgcn_wmma_f32_16x16x4_f32)
  const int wv = threadIdx.x >> 5, lane = threadIdx.x & 31;
  const int half = lane >> 4, ln = lane & 15;
  for (int dt = wv; dt < DG_ / 16; dt += 4) {
    int d = dt * 16 + ln;
    int ch = d / 100; int rem = d - ch * 100;
    int pi = rem / 10; int pj = rem - pi * 10;
    // gather base for this lane's column: x[b, ch, wys*5 + cy*5 + pi, wxs*5 + cx*5 + pj]
    const float* xd = xb + (size_t)ch * HW_ + (size_t)(wys * S_ + pi) * W_ + wxs * S_ + pj;
    v8f acc = {};
    for (int kt = 0; kt < 57; ++kt) {          // K = 228 (225 + 3 zero-padded)
      int c0 = kt * 4 + 2 * half;
      bool kok = (ln < K_);
      v2f a;
      a.x = (kok && c0     < O2_) ? sw[ln * O2_ + c0]     : 0.f;
      a.y = (kok && c0 + 1 < O2_) ? sw[ln * O2_ + c0 + 1] : 0.f;
      int cA = min(c0, O2_ - 1), cB = min(c0 + 1, O2_ - 1);
      int cyA = cA / O_, cxA = cA - cyA * O_;
      int cyB = cB / O_, cxB = cB - cyB * O_;
      v2f bf;
      bf.x = xd[(size_t)(cyA * S_) * W_ + cxA * S_];
      bf.y = xd[(size_t)(cyB * S_) * W_ + cxB * S_];
      acc = __builtin_amdgcn_wmma_f32_16x16x4_f32(
          false, a, false, bf, (short)0, acc, false, false);
    }
    // scatter: element (r, lane) -> (kk = r + 8*half, d); only kk < 7 real
    if (half == 0) {
      float* dst = ob + (size_t)ch * HW_ + (size_t)(gy * S_ + pi) * W_ + gx * S_ + pj;
      #pragma unroll
      for (int r = 0; r < K_; ++r)
        atomicAdd(dst + (size_t)(r * C_) * HW_, acc[r]);
    }
  }
#else
  // host-parse / no-builtin fallback: direct weighted gather + atomic scatter
  for (int e = threadIdx.x; e < K_ * DG_; e += 128) {
    int kk = e / DG_, d = e % DG_;
    int ch = d / 100, rem = d % 100, pi = rem / 10, pj = rem % 10;
    float acc = 0.f;
    for (int c = 0; c < O2_; ++c) {
      int cy = c / O_, cx = c % O_;
      acc += sw[kk * O2_ + c] *
             xb[(size_t)ch * HW_ + (size_t)((wys + cy) * S_ + pi) * W_ + (wxs + cx) * S_ + pj];
    }
    atomicAdd(ob + (size_t)(kk * C_ + ch) * HW_ + (size_t)(gy * S_ + pi) * W_ + gx * S_ + pj, acc);
  }
#endif
}

// ---------------------------------------------------------------------------
// Divide the aggregated channels by the fold coverage count.
// ---------------------------------------------------------------------------
__global__ __launch_bounds__(256)
void k_norm(float* __restrict__ out)
{
  int idx = blockIdx.x * 256 + threadIdx.x;
  if (idx >= B_ * K_ * C_ * HW_) return;
  int px = idx % W_, py = (idx / W_) % H_;
  int cc = (idx / HW_) % (K_ * C_);
  int b = idx / (K_ * C_ * HW_);
  int gylo = (py >= P_ - 1) ? (py - P_ + 1 + S_ - 1) / S_ : 0;
  int gyhi = min(py / S_, G_ - 1);
  int gxlo = (px >= P_ - 1) ? (px - P_ + 1 + S_ - 1) / S_ : 0;
  int gxhi = min(px / S_, G_ - 1);
  float cnt = (float)((gyhi - gylo + 1) * (gxhi - gxlo + 1));
  size_t o = (size_t)(b * (C_ + K_ * C_) + C_ + cc) * HW_ + (size_t)py * W_ + px;
  out[o] = out[o] / cnt;
}

// ---------------------------------------------------------------------------
// Host orchestration. Workspace (~127 MB of floats):
//   buf1 [B*F*HW] -> reused as logits [B*M*225]
//   buf2 [B*F*HW] -> reused as Wk     [B*M*K*225]
//   xe [B*C*HW], lt [B*HW], ynorm [B*M], ltm [B*M], wpack [36864]
// ---------------------------------------------------------------------------
extern "C" void kernel_launch(void* const* d_in, const int* in_sizes, int n_in,
                              void* d_out, int out_size, void* d_ws, size_t ws_size,
                              hipStream_t stream)
{
  (void)in_sizes; (void)n_in; (void)ws_size;
  const float* x    = (const float*)d_in[0];
  const float* e_w1 = (const float*)d_in[1];
  const float* e_g1 = (const float*)d_in[2];
  const float* e_b1 = (const float*)d_in[3];
  const float* e_m1 = (const float*)d_in[4];
  const float* e_v1 = (const float*)d_in[5];
  const float* e_w2 = (const float*)d_in[6];
  const float* e_g2 = (const float*)d_in[7];
  const float* e_b2 = (const float*)d_in[8];
  const float* e_m2 = (const float*)d_in[9];
  const float* e_v2 = (const float*)d_in[10];
  const float* e_w3 = (const float*)d_in[11];
  const float* e_c3 = (const float*)d_in[12];
  const float* t_w1 = (const float*)d_in[13];
  const float* t_g1 = (const float*)d_in[14];
  const float* t_b1 = (const float*)d_in[15];
  const float* t_m1 = (const float*)d_in[16];
  const float* t_v1 = (const float*)d_in[17];
  const float* t_w2 = (const float*)d_in[18];
  const float* t_g2 = (const float*)d_in[19];
  const float* t_b2 = (const float*)d_in[20];
  const float* t_m2 = (const float*)d_in[21];
  const float* t_v2 = (const float*)d_in[22];
  const float* t_w3 = (const float*)d_in[23];
  const float* t_c3 = (const float*)d_in[24];

  float* ws = (float*)d_ws;
  const size_t NBUF = (size_t)B_ * F_ * HW_;
  float* buf1  = ws;
  float* buf2  = buf1 + NBUF;
  float* xe    = buf2 + NBUF;
  float* ltb   = xe + (size_t)B_ * C_ * HW_;
  float* ynorm = ltb + (size_t)B_ * HW_;
  float* ltm   = ynorm + (size_t)B_ * M_;
  float* wpk   = ltm + (size_t)B_ * M_;
  float* logits = buf1;   // free after both CNNs
  float* Wk     = buf2;   // free after conv3
  float* outp = (float*)d_out;

  const int g_conv1 = (B_ * F_ * HW_ + 255) / 256;
  const int g_pack  = (9 * 16 * 4 * 64 + 255) / 256;
  const int g_conv2 = B_ * H_ * (W_ / 16);
  const int g_c3e   = (B_ * C_ * HW_ + 255) / 256;
  const int g_c3t   = (B_ * 1 * HW_ + 255) / 256;
  const int g_wav   = (B_ * M_ * 32 + 255) / 256;
  const int g_dist  = B_ * G_ * 3 * O_;        // (b, gy, gx-tile, cy)
  const int g_copy  = (B_ * C_ * HW_ + 255) / 256;
  const int g_norm  = (B_ * K_ * C_ * HW_ + 255) / 256;

  hipMemsetAsync(d_out, 0, (size_t)out_size * sizeof(float), stream);

  // --- embedding CNN ---
  k_conv1<<<g_conv1, 256, 0, stream>>>(x, e_w1, e_g1, e_b1, e_m1, e_v1, buf1);
  k_packw<<<g_pack, 256, 0, stream>>>(e_w2, wpk);
  k_conv2<<<g_conv2, 128, 0, stream>>>(buf1, e_w2, wpk, e_g2, e_b2, e_m2, e_v2, buf2);
  k_conv3<<<g_c3e, 256, 0, stream>>>(buf2, e_w3, e_c3, xe, C_);
  // --- temperature CNN ---
  k_conv1<<<g_conv1, 256, 0, stream>>>(x, t_w1, t_g1, t_b1, t_m1, t_v1, buf1);
  k_packw<<<g_pack, 256, 0, stream>>>(t_w2, wpk);
  k_conv2<<<g_conv2, 128, 0, stream>>>(buf1, t_w2, wpk, t_g2, t_b2, t_m2, t_v2, buf2);
  k_conv3<<<g_c3t, 256, 0, stream>>>(buf2, t_w3, t_c3, ltb, 1);
  // --- stats, banded-correlation distances (WMMA), softmax chain ---
  k_stats<<<g_wav, 256, 0, stream>>>(xe, ltb, ynorm, ltm);
  k_distw<<<g_dist, 64, 0, stream>>>(xe, ynorm, ltm, logits);
  k_softmax<<<g_wav, 256, 0, stream>>>(logits, Wk);
  // --- output: passthrough + WMMA aggregation (atomic fold) + normalize ---
  k_copyx<<<g_copy, 256, 0, stream>>>(x, outp);
  k_aggw<<<B_ * M_, 128, 0, stream>>>(x, Wk, outp);
  k_norm<<<g_norm, 256, 0, stream>>>(outp);
}